// MambaModel_27728308863788
// MI455X (gfx1250) — compile-verified
//
#include <hip/hip_runtime.h>
#include <math.h>

#define D_MODEL 280
#define D_STATE 16
#define D_CONV  4
#define D_INNER 560
#define DT_RANK 18
#define N_LAYERS 3
#define BATCH 8
#define SEQ 4096
#define M_TOTAL (BATCH * SEQ)           /* 32768 rows */
#define PROJ_W  (DT_RANK + 2 * D_STATE) /* 50 */
#define CHUNK 128
#define NCHUNK (SEQ / CHUNK)            /* 32 */
#define CPAIRS (D_INNER / 2)            /* 280: 2 channels per wave32 */

typedef float v2f __attribute__((ext_vector_type(2)));
typedef float v8f __attribute__((ext_vector_type(8)));

__device__ __forceinline__ float sigmoidf_(float x) { return 1.0f / (1.0f + __expf(-x)); }

// ---------------------------------------------------------------------------
// C[M x Nreal] = A[M x K] * W[Nreal x K]^T
// K must be a multiple of 8 (true for 280 and 560). No guards in the hot loop:
// out-of-range weight rows are clamped to a valid row (garbage lands in output
// columns that are never stored). One wave owns TWO 16x16 tiles (shared A-frag),
// K unrolled by 2 so loads overlap WMMAs. 4 waves/block -> 8 n-tiles per block.
// ---------------------------------------------------------------------------
__global__ __launch_bounds__(128)
void gemm_xt(const float* __restrict__ A, int lda,
             const float* __restrict__ W, int ldw,
             float* __restrict__ C, int ldc,
             int Nreal, int K)
{
    int lane = threadIdx.x & 31;
    int wv   = threadIdx.x >> 5;
    int m0   = blockIdx.x << 4;
    int ntiles = (Nreal + 15) >> 4;
    int nt0  = (blockIdx.y * 4 + wv) * 2;
    if (nt0 >= ntiles) return;            // wave-uniform exit

    int r  = lane & 15;
    int kh = (lane >> 4) << 1;            // lanes 0-15: K pair {k,k+1}; 16-31: {k+2,k+3}

    const float* arow = A + (size_t)(m0 + r) * lda + kh;
    int wr0 = nt0 * 16 + r;       if (wr0 >= Nreal) wr0 = Nreal - 1;
    int wr1 = nt0 * 16 + 16 + r;  if (wr1 >= Nreal) wr1 = Nreal - 1;
    const float* brow0 = W + (size_t)wr0 * ldw + kh;
    const float* brow1 = W + (size_t)wr1 * ldw + kh;

    v8f acc0 = {}, acc1 = {};
    for (int k = 0; k < K; k += 8) {
        v2f a0  = *(const v2f*)(arow  + k);
        v2f a1  = *(const v2f*)(arow  + k + 4);
        v2f b00 = *(const v2f*)(brow0 + k);
        v2f b01 = *(const v2f*)(brow0 + k + 4);
        v2f b10 = *(const v2f*)(brow1 + k);
        v2f b11 = *(const v2f*)(brow1 + k + 4);
        acc0 = __builtin_amdgcn_wmma_f32_16x16x4_f32(false, a0, false, b00, (short)0, acc0, false, false);
        acc1 = __builtin_amdgcn_wmma_f32_16x16x4_f32(false, a0, false, b10, (short)0, acc1, false, false);
        acc0 = __builtin_amdgcn_wmma_f32_16x16x4_f32(false, a1, false, b01, (short)0, acc0, false, false);
        acc1 = __builtin_amdgcn_wmma_f32_16x16x4_f32(false, a1, false, b11, (short)0, acc1, false, false);
    }

    // D layout: VGPR i -> row m0+i (lanes 0-15) / m0+8+i (lanes 16-31), col = lane&15
    int n  = lane & 15;
    int mb = m0 + ((lane >> 4) << 3);
    int na = nt0 * 16 + n;
    int nb = nt0 * 16 + 16 + n;
    if (na < Nreal) {
        #pragma unroll
        for (int i = 0; i < 8; ++i) C[(size_t)(mb + i) * ldc + na] = acc0[i];
    }
    if (nb < Nreal) {
        #pragma unroll
        for (int i = 0; i < 8; ++i) C[(size_t)(mb + i) * ldc + nb] = acc1[i];
    }
}

// ---------------------------------------------------------------------------
// dt_proj (K=18) + bias + softplus as a plain VALU kernel: proj rows broadcast
// across the 560 channel-lanes of each token; dtw (40 KB) is L1/L2 resident.
// ---------------------------------------------------------------------------
__global__ void dtproj_softplus_k(const float* __restrict__ proj,
                                  const float* __restrict__ dtw,
                                  const float* __restrict__ bias,
                                  float* __restrict__ dtb)
{
    int idx = blockIdx.x * blockDim.x + threadIdx.x;
    if (idx >= M_TOTAL * D_INNER) return;
    int c  = idx % D_INNER;
    int bt = idx / D_INNER;
    const float* pr = proj + (size_t)bt * PROJ_W;
    const float* wr = dtw + c * DT_RANK;
    float acc = bias[c];
    #pragma unroll
    for (int k = 0; k < DT_RANK; ++k) acc = fmaf(pr[k], wr[k], acc);
    dtb[(size_t)idx] = (acc > 20.0f) ? acc : log1pf(__expf(acc));
}

// ---------------------------------------------------------------------------
// Depthwise causal conv (K=4) + bias + SiLU. Reads xin = xz[..., :560].
// ---------------------------------------------------------------------------
__global__ void conv_silu_k(const float* __restrict__ xz,
                            const float* __restrict__ cw,
                            const float* __restrict__ cb,
                            float* __restrict__ xc)
{
    int idx = blockIdx.x * blockDim.x + threadIdx.x;
    if (idx >= M_TOTAL * D_INNER) return;
    int c  = idx % D_INNER;
    int bt = idx / D_INNER;                 // b*SEQ + t
    int t  = bt % SEQ;
    float acc = cb[c];
    #pragma unroll
    for (int k = 0; k < D_CONV; ++k) {
        int tt = t - (D_CONV - 1) + k;
        if (tt >= 0)
            acc += xz[(size_t)(bt - (D_CONV - 1) + k) * (2 * D_INNER) + c] * cw[c * D_CONV + k];
    }
    xc[(size_t)bt * D_INNER + c] = acc * sigmoidf_(acc);
}

// ---------------------------------------------------------------------------
// Chunked scan, phase 1: per (b, channel-pair, chunk) wave, run h from 0 over
// CHUNK steps; record final h and product of dA per (c, n) lane.
// Lane = (chan_half << 4) | n.
// ---------------------------------------------------------------------------
__global__ void scan_phase1(const float* __restrict__ dtb,
                            const float* __restrict__ proj,
                            const float* __restrict__ xc,
                            const float* __restrict__ Alog,
                            float* __restrict__ hseg,
                            float* __restrict__ aprod)
{
    int wid  = (blockIdx.x * blockDim.x + threadIdx.x) >> 5;
    int lane = threadIdx.x & 31;
    int chunk = wid & (NCHUNK - 1);
    int cp    = (wid / NCHUNK) % CPAIRS;
    int b     = wid / (NCHUNK * CPAIRS);
    int n = lane & 15;
    int c = cp * 2 + (lane >> 4);

    float Aval = -__expf(Alog[c * D_STATE + n]);
    size_t bt0 = (size_t)b * SEQ + (size_t)chunk * CHUNK;
    const float* dtp = dtb  + bt0 * D_INNER + c;
    const float* up  = xc   + bt0 * D_INNER + c;
    const float* Bp  = proj + bt0 * PROJ_W + DT_RANK + n;

    float h = 0.0f, ap = 1.0f;
    for (int i = 0; i < CHUNK; ++i) {
        float dt = *dtp;
        float u  = *up;
        float Bv = *Bp;
        float dA = __expf(dt * Aval);
        h  = dA * h + (dt * u) * Bv;
        ap *= dA;
        dtp += D_INNER; up += D_INNER; Bp += PROJ_W;
    }
    size_t o = ((size_t)(b * CPAIRS + cp) * NCHUNK + chunk) * 32 + lane;
    hseg[o]  = h;
    aprod[o] = ap;
}

// Phase 2: 32-step serial prefix across chunks per (b, c, n).
__global__ void scan_phase2(const float* __restrict__ hseg,
                            const float* __restrict__ aprod,
                            float* __restrict__ hinit)
{
    int tid = blockIdx.x * blockDim.x + threadIdx.x;
    if (tid >= BATCH * CPAIRS * 32) return;
    int lane = tid & 31;
    int bcp  = tid >> 5;
    size_t base = (size_t)bcp * NCHUNK * 32 + lane;
    float h = 0.0f;
    for (int k = 0; k < NCHUNK; ++k) {
        size_t o = base + (size_t)k * 32;
        hinit[o] = h;
        h = aprod[o] * h + hseg[o];
    }
}

// Phase 3: re-run each chunk from corrected h; y = sum_n h*C (shfl_xor reduce),
// fused with +D*u and silu(z) gate; y overwrites u (xc) in place.
__global__ void scan_phase3(const float* __restrict__ dtb,
                            const float* __restrict__ proj,
                            const float* __restrict__ xz,      // z = xz[..., 560:]
                            const float* __restrict__ Alog,
                            const float* __restrict__ Dvec,
                            const float* __restrict__ hinit,
                            float* __restrict__ xc)            // u in / y out
{
    int wid  = (blockIdx.x * blockDim.x + threadIdx.x) >> 5;
    int lane = threadIdx.x & 31;
    int chunk = wid & (NCHUNK - 1);
    int cp    = (wid / NCHUNK) % CPAIRS;
    int b     = wid / (NCHUNK * CPAIRS);
    int n = lane & 15;
    int c = cp * 2 + (lane >> 4);

    float Aval = -__expf(Alog[c * D_STATE + n]);
    float Dc   = Dvec[c];
    size_t bt0 = (size_t)b * SEQ + (size_t)chunk * CHUNK;
    const float* dtp = dtb  + bt0 * D_INNER + c;
    float*       up  = xc   + bt0 * D_INNER + c;
    const float* Bp  = proj + bt0 * PROJ_W + DT_RANK + n;
    const float* Cp  = proj + bt0 * PROJ_W + DT_RANK + D_STATE + n;
    const float* zp  = xz   + bt0 * (2 * D_INNER) + D_INNER + c;

    float h = hinit[((size_t)(b * CPAIRS + cp) * NCHUNK + chunk) * 32 + lane];
    for (int i = 0; i < CHUNK; ++i) {
        float dt = *dtp;
        float u  = *up;
        float Bv = *Bp;
        float Cv = *Cp;
        float dA = __expf(dt * Aval);
        h = dA * h + (dt * u) * Bv;
        float y = h * Cv;
        y += __shfl_xor(y, 1, 32);
        y += __shfl_xor(y, 2, 32);
        y += __shfl_xor(y, 4, 32);
        y += __shfl_xor(y, 8, 32);
        if (n == 0) {                      // lanes 0 and 16: one per channel
            float z = *zp;
            *up = (y + Dc * u) * (z * sigmoidf_(z));
        }
        dtp += D_INNER; up += D_INNER; Bp += PROJ_W; Cp += PROJ_W; zp += 2 * D_INNER;
    }
}

// ---------------------------------------------------------------------------
extern "C" void kernel_launch(void* const* d_in, const int* in_sizes, int n_in,
                              void* d_out, int out_size, void* d_ws, size_t ws_size,
                              hipStream_t stream)
{
    const float* x_in   = (const float*)d_in[0];
    const float* inw    = (const float*)d_in[1];
    const float* cw     = (const float*)d_in[2];
    const float* cb     = (const float*)d_in[3];
    const float* xpw    = (const float*)d_in[4];
    const float* dtw    = (const float*)d_in[5];
    const float* dtbias = (const float*)d_in[6];
    const float* Alog   = (const float*)d_in[7];
    const float* Dv     = (const float*)d_in[8];
    const float* ow     = (const float*)d_in[9];
    float* out = (float*)d_out;

    float* ws = (float*)d_ws;
    size_t off = 0;
    float* xz    = ws + off; off += (size_t)M_TOTAL * 2 * D_INNER;   // 146.8 MB
    float* xc    = ws + off; off += (size_t)M_TOTAL * D_INNER;       // 73.4 MB
    float* proj  = ws + off; off += (size_t)M_TOTAL * PROJ_W;        // 6.6 MB
    float* dtb   = ws + off; off += (size_t)M_TOTAL * D_INNER;       // 73.4 MB
    float* xbuf  = ws + off; off += (size_t)M_TOTAL * D_MODEL;       // 36.7 MB
    size_t segN  = (size_t)BATCH * CPAIRS * NCHUNK * 32;             // 2.29 M
    float* hseg  = ws + off; off += segN;
    float* aprod = ws + off; off += segN;
    float* hinit = ws + off; off += segN;
    (void)ws_size; (void)in_sizes; (void)n_in; (void)out_size;

    dim3 blk(128);
    const int nwaves = BATCH * CPAIRS * NCHUNK;                      // 71680
    const int elems  = M_TOTAL * D_INNER;                            // 18.35 M

    for (int l = 0; l < N_LAYERS; ++l) {
        const float* xcur = (l == 0) ? x_in : xbuf;

        // in_proj: xz = x @ W_in^T  (N=1120 -> 70 tiles -> grid.y=9, K=280)
        gemm_xt<<<dim3(M_TOTAL / 16, 9), blk, 0, stream>>>(
            xcur, D_MODEL, inw + (size_t)l * 2 * D_INNER * D_MODEL, D_MODEL,
            xz, 2 * D_INNER, 2 * D_INNER, D_MODEL);

        // depthwise causal conv + SiLU -> xc
        conv_silu_k<<<(elems + 255) / 256, 256, 0, stream>>>(
            xz, cw + (size_t)l * D_INNER * D_CONV, cb + (size_t)l * D_INNER, xc);

        // x_proj: proj = xc @ xpw^T  (N=50 -> 4 tiles -> grid.y=1, K=560)
        gemm_xt<<<dim3(M_TOTAL / 16, 1), blk, 0, stream>>>(
            xc, D_INNER, xpw + (size_t)l * PROJ_W * D_INNER, D_INNER,
            proj, PROJ_W, PROJ_W, D_INNER);

        // dt_proj + softplus (K=18, VALU kernel)
        dtproj_softplus_k<<<(elems + 255) / 256, 256, 0, stream>>>(
            proj, dtw + (size_t)l * D_INNER * DT_RANK,
            dtbias + (size_t)l * D_INNER, dtb);

        // chunked selective scan (3 phases), gate fused into phase 3
        const float* Al = Alog + (size_t)l * D_INNER * D_STATE;
        scan_phase1<<<nwaves / 4, blk, 0, stream>>>(dtb, proj, xc, Al, hseg, aprod);
        scan_phase2<<<(BATCH * CPAIRS * 32 + 255) / 256, 256, 0, stream>>>(hseg, aprod, hinit);
        scan_phase3<<<nwaves / 4, blk, 0, stream>>>(dtb, proj, xz, Al,
                                                    Dv + (size_t)l * D_INNER, hinit, xc);

        // out_proj: dst = y @ ow^T  (N=280 -> 18 tiles -> grid.y=3, K=560)
        float* dst = (l == N_LAYERS - 1) ? out : xbuf;
        gemm_xt<<<dim3(M_TOTAL / 16, 3), blk, 0, stream>>>(
            xc, D_INNER, ow + (size_t)l * D_MODEL * D_INNER, D_INNER,
            dst, D_MODEL, D_MODEL, D_INNER);
    }
}